// GNN_7653631722064
// MI455X (gfx1250) — compile-verified
//
#include <hip/hip_runtime.h>
#include <hip/hip_bf16.h>
#include <limits.h>

// Problem constants (match reference)
#define Nn      10000
#define Ee      120000
#define IN_DIM  256
#define Hh      8
#define Cc      128
#define Gg      64
#define HC      1024          // Hh*Cc
#define Et      (Ee + Nn)     // edges + self loops
#define EPSbn   1e-5f
#define SLOPE   0.2f

typedef float v2f __attribute__((ext_vector_type(2)));
typedef float v8f __attribute__((ext_vector_type(8)));
typedef int   v4i __attribute__((ext_vector_type(4)));

__device__ __forceinline__ float lrelu(float z) { return z > 0.f ? z : SLOPE * z; }

// IEEE-monotonic float atomic max (init with large negative value)
__device__ __forceinline__ void atomicMaxF(float* addr, float val) {
  if (val >= 0.f) atomicMax((int*)addr, __float_as_int(val));
  else            atomicMin((unsigned int*)addr, __float_as_uint(val));
}

// ---------------------------------------------------------------------------
// Global -> LDS 16B copy: CDNA5 async-to-LDS (ASYNCcnt) when available,
// synchronous float4 staging otherwise. Builtin prototype (from hipcc
// diagnostic): param0 = v4i in AS(1) [global], then LDS dst, imm offset, cpol.
// ---------------------------------------------------------------------------
#if defined(__gfx1250__) && __has_builtin(__builtin_amdgcn_global_load_async_to_lds_b128)
#define USE_ASYNC_LDS 1
#endif

__device__ __forceinline__ void cp_f4_to_lds(float* lds_dst, const float* gsrc) {
#ifdef USE_ASYNC_LDS
  __builtin_amdgcn_global_load_async_to_lds_b128(
      (__attribute__((address_space(1))) v4i*)gsrc,
      (__attribute__((address_space(3))) v4i*)lds_dst, 0, 0);
#else
  *(float4*)lds_dst = *(const float4*)gsrc;
#endif
}

__device__ __forceinline__ void cp_lds_wait() {
#ifdef USE_ASYNC_LDS
#if __has_builtin(__builtin_amdgcn_s_wait_asynccnt)
  __builtin_amdgcn_s_wait_asynccnt(0);
#else
  asm volatile("s_wait_asynccnt 0" ::: "memory");
#endif
#endif
}

// ---------------------------------------------------------------------------
// C[M,1024] = A[M,K] @ B[K,1024], f32 WMMA 16x16x4 (signature verified rd 1).
// Block = 256 thr (8 waves). Block tile M=16 x N=128; wave w owns cols
// nb + w*16 .. +15. A slab (16xK) staged once; B staged in 32-row K-chunks.
// All inner-loop operands come from padded, conflict-free LDS.
// M%16==0 (10000=625*16), K%32==0 (256/128).
// ---------------------------------------------------------------------------
#define LDBS 144                        // 32-row B chunk, padded row stride
__global__ void gemm_wmma_f32(const float* __restrict__ A,
                              const float* __restrict__ B,
                              float* __restrict__ C, int K) {
  extern __shared__ float lds[];
  const int ldAs = K + 4;               // lane banks {4r} / {4r+2}: disjoint
  float* As = lds;                      // 16 * ldAs
  float* Bs = lds + 16 * ldAs;          // 32 * LDBS
  const int tid = threadIdx.x;
  const int m0  = blockIdx.y * 16;
  const int nb  = blockIdx.x * 128;
  const int K4  = K >> 2;

  // stage A slab (16 x K) as b128 copies, coalesced along rows
  for (int idx4 = tid; idx4 < 16 * K4; idx4 += 256) {
    const int row = idx4 / K4, c4 = idx4 % K4;
    cp_f4_to_lds(&As[row * ldAs + c4 * 4], &A[(m0 + row) * K + c4 * 4]);
  }

  const int wave = tid >> 5, lane = tid & 31;
  const int hf = lane >> 4, r = lane & 15;   // lanes 0-15: K pair 0; 16-31: K pair 2
  const int nn = wave * 16;

  v8f acc = {};
  for (int kb = 0; kb < K; kb += 32) {
    // stage B chunk rows kb..kb+31, cols nb..nb+127 (32x128 = 1024 float4)
    for (int idx4 = tid; idx4 < 1024; idx4 += 256) {
      const int row = idx4 >> 5, c4 = idx4 & 31;
      cp_f4_to_lds(&Bs[row * LDBS + c4 * 4], &B[(kb + row) * HC + nb + c4 * 4]);
    }
    cp_lds_wait();
    __syncthreads();
#if defined(__gfx1250__) && __has_builtin(__builtin_amdgcn_wmma_f32_16x16x4_f32)
    #pragma unroll
    for (int k = 0; k < 32; k += 4) {
      const int ka = k + 2 * hf;
      v2f a, b;
      a.x = As[r * ldAs + kb + ka];
      a.y = As[r * ldAs + kb + ka + 1];
      b.x = Bs[ka * LDBS + nn + r];
      b.y = Bs[(ka + 1) * LDBS + nn + r];
      acc = __builtin_amdgcn_wmma_f32_16x16x4_f32(false, a, false, b,
                                                  (short)0, acc, false, false);
    }
#else
    for (int i = 0; i < 8; ++i) {       // scalar fallback (host / non-gfx1250 pass)
      float s = acc[i];
      const int row = i + 8 * hf;
      for (int k = 0; k < 32; ++k)
        s += As[row * ldAs + kb + k] * Bs[k * LDBS + nn + r];
      acc[i] = s;
    }
#endif
    __syncthreads();
  }
  #pragma unroll
  for (int i = 0; i < 8; ++i)           // D layout: VGPR i -> row i (+8 hi half)
    C[(m0 + i + 8 * hf) * HC + nb + nn + r] = acc[i];
}

// e_s/e_d: one wave32 per (node, head) dot product over C=128
__global__ void gat_scores(const float* __restrict__ h,
                           const float* __restrict__ a_s,
                           const float* __restrict__ a_d,
                           float* __restrict__ es, float* __restrict__ ed) {
  const int w = (blockIdx.x * blockDim.x + threadIdx.x) >> 5;
  const int lane = threadIdx.x & 31;
  if (w >= Nn * Hh) return;
  const int n = w >> 3, hd = w & 7;
  const float* hr  = h + n * HC + hd * Cc;
  const float* asr = a_s + hd * Cc;
  const float* adr = a_d + hd * Cc;
  float ss = 0.f, sd = 0.f;
  #pragma unroll
  for (int j = lane; j < Cc; j += 32) {
    const float v = hr[j];
    ss += v * asr[j];
    sd += v * adr[j];
  }
  #pragma unroll
  for (int off = 16; off > 0; off >>= 1) {
    ss += __shfl_xor(ss, off, 32);
    sd += __shfl_xor(sd, off, 32);
  }
  if (lane == 0) { es[w] = ss; ed[w] = sd; }
}

__global__ void init_layer(float* __restrict__ accb, float* __restrict__ mx,
                           float* __restrict__ den) {
  const int gid = blockIdx.x * blockDim.x + threadIdx.x;   // covers Nn*HC exactly
  accb[gid] = 0.f;
  if (gid < Nn * Hh) { mx[gid] = -3.0e38f; den[gid] = 0.f; }
}

__device__ __forceinline__ void edge_sd(int e, const int* __restrict__ ei, int& s, int& d) {
  if (e < Ee) { s = ei[e]; d = ei[Ee + e]; } else { s = d = e - Ee; }
}

__global__ void edge_max(const int* __restrict__ ei, const float* __restrict__ es,
                         const float* __restrict__ ed, float* __restrict__ mx) {
  const int e = blockIdx.x * blockDim.x + threadIdx.x;
  if (e >= Et) return;
  int s, d; edge_sd(e, ei, s, d);
  #pragma unroll
  for (int hd = 0; hd < Hh; ++hd)
    atomicMaxF(&mx[d * Hh + hd], lrelu(es[s * Hh + hd] + ed[d * Hh + hd]));
}

__global__ void edge_sum(const int* __restrict__ ei, const float* __restrict__ es,
                         const float* __restrict__ ed, const float* __restrict__ mx,
                         float* __restrict__ den) {
  const int e = blockIdx.x * blockDim.x + threadIdx.x;
  if (e >= Et) return;
  int s, d; edge_sd(e, ei, s, d);
  #pragma unroll
  for (int hd = 0; hd < Hh; ++hd) {
    const float al = lrelu(es[s * Hh + hd] + ed[d * Hh + hd]);
    atomicAdd(&den[d * Hh + hd], __expf(al - mx[d * Hh + hd]));
  }
}

// one block per edge, wave w handles head w; atomics land in L2 (h is L2-resident)
__global__ void edge_scatter(const int* __restrict__ ei, const float* __restrict__ es,
                             const float* __restrict__ ed, const float* __restrict__ mx,
                             const float* __restrict__ den, const float* __restrict__ h,
                             float* __restrict__ accb) {
  const int e = blockIdx.x;
  const int hd = threadIdx.x >> 5, lane = threadIdx.x & 31;
  int s, d; edge_sd(e, ei, s, d);
  const float al   = lrelu(es[s * Hh + hd] + ed[d * Hh + hd]);
  const float coef = __expf(al - mx[d * Hh + hd]) / den[d * Hh + hd];
  const float* hr = h + s * HC + hd * Cc;
  float* orow = accb + d * HC + hd * Cc;
  #pragma unroll
  for (int j = lane; j < Cc; j += 32)
    atomicAdd(&orow[j], hr[j] * coef);
}

// head-mean + bias + ReLU + BatchNorm(eval)
__global__ void node_finalize(const float* __restrict__ accb, const float* __restrict__ bias,
                              const float* __restrict__ gam, const float* __restrict__ bet,
                              const float* __restrict__ mu, const float* __restrict__ var,
                              float* __restrict__ xout) {
  const int gid = blockIdx.x * blockDim.x + threadIdx.x;   // Nn*Cc exactly
  const int n = gid >> 7, c = gid & 127;
  float s = 0.f;
  #pragma unroll
  for (int hd = 0; hd < Hh; ++hd) s += accb[n * HC + hd * Cc + c];
  float z = s * (1.f / Hh) + bias[c];
  z = fmaxf(z, 0.f);
  z = (z - mu[c]) * rsqrtf(var[c] + EPSbn) * gam[c] + bet[c];
  xout[n * Cc + c] = z;
}

__global__ void pool_init(float* __restrict__ gmp, float* __restrict__ gsum,
                          float* __restrict__ cnt, int* __restrict__ root) {
  const int gid = blockIdx.x * blockDim.x + threadIdx.x;   // Gg*Cc exactly
  gmp[gid] = -3.0e38f; gsum[gid] = 0.f;
  if (gid < Gg) { cnt[gid] = 0.f; root[gid] = INT_MAX; }
}

__global__ void pool_kernel(const float* __restrict__ xf, const int* __restrict__ batch,
                            float* __restrict__ gmp, float* __restrict__ gsum,
                            float* __restrict__ cnt, int* __restrict__ root) {
  const int gid = blockIdx.x * blockDim.x + threadIdx.x;   // Nn*Cc exactly
  const int n = gid >> 7, c = gid & 127;
  const int b = batch[n];
  const float v = xf[gid];
  atomicMaxF(&gmp[b * Cc + c], v);
  atomicAdd(&gsum[b * Cc + c], v);
  if (c == 0) { atomicAdd(&cnt[b], 1.f); atomicMin(&root[b], n); }
}

// per-graph head: hg = relu([gmp,gap]@l0W+l0b); news = relu(x[root]@lnW+lnb);
// out = sigmoid([hg,news]@l1W + l1b)
__global__ void head_kernel(const float* __restrict__ gmp, const float* __restrict__ gsum,
                            const float* __restrict__ cnt, const int* __restrict__ root,
                            const float* __restrict__ x,
                            const float* __restrict__ lnW, const float* __restrict__ lnb,
                            const float* __restrict__ l0W, const float* __restrict__ l0b,
                            const float* __restrict__ l1W, const float* __restrict__ l1b,
                            float* __restrict__ out) {
  __shared__ float feat[2 * Cc];
  __shared__ float red[Cc];
  const int g = blockIdx.x, c = threadIdx.x;   // 64 blocks x 128 threads
  const float invc = 1.f / cnt[g];
  float acc = l0b[c];
  for (int j = 0; j < Cc; ++j) acc += gmp[g * Cc + j] * l0W[j * Cc + c];
  for (int j = 0; j < Cc; ++j) acc += gsum[g * Cc + j] * invc * l0W[(Cc + j) * Cc + c];
  feat[c] = fmaxf(acc, 0.f);
  const int rt = root[g];
  float a2 = lnb[c];
  for (int j = 0; j < IN_DIM; ++j) a2 += x[rt * IN_DIM + j] * lnW[j * Cc + c];
  feat[Cc + c] = fmaxf(a2, 0.f);
  __syncthreads();
  red[c] = feat[c] * l1W[c] + feat[Cc + c] * l1W[Cc + c];
  __syncthreads();
  for (int s = 64; s > 0; s >>= 1) { if (c < s) red[c] += red[c + s]; __syncthreads(); }
  if (c == 0) out[g] = 1.f / (1.f + __expf(-(red[0] + l1b[0])));
}

// ---------------------------------------------------------------------------
extern "C" void kernel_launch(void* const* d_in, const int* in_sizes, int n_in,
                              void* d_out, int out_size, void* d_ws, size_t ws_size,
                              hipStream_t stream) {
  const float* x     = (const float*)d_in[0];
  const int*   ei    = (const int*)d_in[1];
  const int*   batch = (const int*)d_in[2];
  const float* Wl[3]  = {(const float*)d_in[3],  (const float*)d_in[11], (const float*)d_in[19]};
  const float* Asl[3] = {(const float*)d_in[4],  (const float*)d_in[12], (const float*)d_in[20]};
  const float* Adl[3] = {(const float*)d_in[5],  (const float*)d_in[13], (const float*)d_in[21]};
  const float* Bl[3]  = {(const float*)d_in[6],  (const float*)d_in[14], (const float*)d_in[22]};
  const float* Gl[3]  = {(const float*)d_in[7],  (const float*)d_in[15], (const float*)d_in[23]};
  const float* BBl[3] = {(const float*)d_in[8],  (const float*)d_in[16], (const float*)d_in[24]};
  const float* Ml[3]  = {(const float*)d_in[9],  (const float*)d_in[17], (const float*)d_in[25]};
  const float* Vl[3]  = {(const float*)d_in[10], (const float*)d_in[18], (const float*)d_in[26]};
  const float* lnW = (const float*)d_in[27];
  const float* lnb = (const float*)d_in[28];
  const float* l0W = (const float*)d_in[29];
  const float* l0b = (const float*)d_in[30];
  const float* l1W = (const float*)d_in[31];
  const float* l1b = (const float*)d_in[32];
  float* outp = (float*)d_out;

  // workspace layout (float elements)
  float* ws   = (float*)d_ws;
  float* h    = ws;                       // Nn*HC
  float* accb = ws + (size_t)10240000;    // Nn*HC
  float* xbuf = ws + (size_t)20480000;    // Nn*Cc
  float* es   = ws + (size_t)21760000;    // Nn*Hh
  float* ed   = ws + (size_t)21840000;    // Nn*Hh
  float* mx   = ws + (size_t)21920000;    // Nn*Hh
  float* den  = ws + (size_t)22000000;    // Nn*Hh
  float* gmp  = ws + (size_t)22080000;    // Gg*Cc
  float* gsum = ws + (size_t)22088192;    // Gg*Cc
  float* cnt  = ws + (size_t)22096384;    // Gg
  int*   root = (int*)(ws + (size_t)22096448); // Gg

  const int edgeBlocks = (Et + 255) / 256;

  for (int lay = 0; lay < 3; ++lay) {
    const float* xin = (lay == 0) ? x : xbuf;
    const int K = (lay == 0) ? IN_DIM : Cc;
    const size_t smem = ((size_t)16 * (K + 4) + 32 * LDBS) * sizeof(float);

    init_layer<<<Nn * HC / 256, 256, 0, stream>>>(accb, mx, den);
    gemm_wmma_f32<<<dim3(HC / 128, Nn / 16), 256, smem, stream>>>(xin, Wl[lay], h, K);
    gat_scores<<<Nn * Hh * 32 / 256, 256, 0, stream>>>(h, Asl[lay], Adl[lay], es, ed);
    edge_max<<<edgeBlocks, 256, 0, stream>>>(ei, es, ed, mx);
    edge_sum<<<edgeBlocks, 256, 0, stream>>>(ei, es, ed, mx, den);
    edge_scatter<<<Et, 256, 0, stream>>>(ei, es, ed, mx, den, h, accb);
    node_finalize<<<Nn * Cc / 256, 256, 0, stream>>>(accb, Bl[lay], Gl[lay], BBl[lay],
                                                     Ml[lay], Vl[lay], xbuf);
  }

  pool_init<<<Gg * Cc / 256, 256, 0, stream>>>(gmp, gsum, cnt, root);
  pool_kernel<<<Nn * Cc / 256, 256, 0, stream>>>(xbuf, batch, gmp, gsum, cnt, root);
  head_kernel<<<Gg, Cc, 0, stream>>>(gmp, gsum, cnt, root, x, lnW, lnb, l0W, l0b,
                                     l1W, l1b, outp);
  (void)in_sizes; (void)n_in; (void)out_size; (void)ws_size;
}